// SequenceFeatureTokenizer_4655744549212
// MI455X (gfx1250) — compile-verified
//
#include <hip/hip_runtime.h>

// SequenceFeatureTokenizer for MI455X (gfx1250, wave32).
// Store-bandwidth-bound (~201 MB f32 output). Numerical tokens computed on the
// matrix pipe via chained V_WMMA_F32_16X16X4_F32 (diag(x) * W + (bias+pos));
// categorical tokens are L2-resident gathers; all output stores non-temporal.

typedef float v2f __attribute__((ext_vector_type(2)));
typedef float v4f __attribute__((ext_vector_type(4)));
typedef float v8f __attribute__((ext_vector_type(8)));

#define BB 256
#define SS 128
#define DD 64
#define NNUM 16
#define NCAT 8
#define CARD 1000
#define TOK_PER_S 24
#define TOK_PER_B 3073   // 1 + 128*24

__global__ __launch_bounds__(256) void tokenizer_kernel(
    const float* __restrict__ x_seq,     // [256][128][24]
    const float* __restrict__ cls_token, // [64]
    const float* __restrict__ W,         // [16][64]
    const float* __restrict__ Bias,      // [16][64]
    const float* __restrict__ cat_emb,   // [8][1000][64]
    const float* __restrict__ pos,       // [128][64]
    float* __restrict__ out)             // [256][3073][64]
{
    const int lane = threadIdx.x & 31;
    const int wave = threadIdx.x >> 5;
    const int pair = blockIdx.x * 8 + wave;   // 0 .. 32767 (one wave per (b,s))
    const int b = pair >> 7;                  // / 128
    const int s = pair & 127;

    const int m  = lane & 15;                 // matrix row (feature) for this lane
    const bool hi = lane >= 16;
    const int k0 = hi ? 2 : 0;                // K base for A/B fragments (16x16x4 layout)

    // x value for this lane's numeric feature row (lanes 16-31 mirror 0-15)
    const float xm = x_seq[((size_t)b * SS + s) * TOK_PER_S + m];

    const size_t out_bs = ((size_t)b * TOK_PER_B + 1 + (size_t)s * TOK_PER_S) * DD;
    const float* pos_s = pos + (size_t)s * DD;

    // ---- numerical tokens: D(16x16) = diag(x16) * W(16x16 tile) + (bias + pos) ----
    // diag split along K into four 16x4 slices -> 4 chained WMMAs per d-tile.
    #pragma unroll
    for (int dt = 0; dt < 4; ++dt) {
        const int dcol = 16 * dt + m;         // output column (d dimension)
        const float posv = pos_s[dcol];

        v8f c;
        #pragma unroll
        for (int r = 0; r < 8; ++r) {
            const int fm = r + (hi ? 8 : 0);  // C/D layout: VGPR r -> M = r (+8 for hi half)
            c[r] = Bias[fm * DD + dcol] + posv;
        }

        #pragma unroll
        for (int j = 0; j < 4; ++j) {
            // A fragment (16x4): slice j of diag(x). Lane holds (M=m, K=k0) and (M=m, K=k0+1).
            v2f a;
            a.x = (m == 4 * j + k0    ) ? xm : 0.0f;
            a.y = (m == 4 * j + k0 + 1) ? xm : 0.0f;
            // B fragment (4x16): rows 4j+k0, 4j+k0+1 of W at column dcol.
            v2f bb;
            bb.x = W[(4 * j + k0    ) * DD + dcol];
            bb.y = W[(4 * j + k0 + 1) * DD + dcol];
            // 8 args: (neg_a, A, neg_b, B, c_mod, C, reuse_a, reuse_b)
            c = __builtin_amdgcn_wmma_f32_16x16x4_f32(
                    false, a, false, bb, (short)0, c, false, false);
        }

        // store 16x16 tile: VGPR r -> row fm, lanes give column dcol (64B segments)
        #pragma unroll
        for (int r = 0; r < 8; ++r) {
            const int fm = r + (hi ? 8 : 0);
            __builtin_nontemporal_store(c[r], &out[out_bs + (size_t)fm * DD + dcol]);
        }
    }

    // ---- categorical tokens: 8 gathers of 64 f32, +pos, float4 NT stores ----
    const float* xc = &x_seq[((size_t)b * SS + s) * TOK_PER_S + NNUM];
    #pragma unroll
    for (int it = 0; it < 4; ++it) {
        const int flat = it * 32 + lane;      // 0..127 float4 slots
        const int tok  = flat >> 4;           // 0..7
        const int q    = flat & 15;           // float4 index within token
        const int idx  = (int)xc[tok];        // trunc-toward-zero matches astype(int32)

        v4f v = *(const v4f*)&cat_emb[(((size_t)tok * CARD) + idx) * DD + q * 4];
        v4f p = *(const v4f*)&pos_s[q * 4];
        v += p;
        __builtin_nontemporal_store(v,
            (v4f*)&out[out_bs + (size_t)(NNUM + tok) * DD + q * 4]);
    }

    // ---- CLS token: one wave per batch (s==0) writes out[b,0,:] ----
    if (s == 0) {
        const size_t base = (size_t)b * TOK_PER_B * DD;
        __builtin_nontemporal_store(cls_token[lane],      &out[base + lane]);
        __builtin_nontemporal_store(cls_token[lane + 32], &out[base + lane + 32]);
    }
}

extern "C" void kernel_launch(void* const* d_in, const int* in_sizes, int n_in,
                              void* d_out, int out_size, void* d_ws, size_t ws_size,
                              hipStream_t stream) {
    const float* x_seq     = (const float*)d_in[0];
    const float* cls_token = (const float*)d_in[1];
    const float* W         = (const float*)d_in[2];
    const float* Bias      = (const float*)d_in[3];
    const float* cat_emb   = (const float*)d_in[4];
    const float* pos       = (const float*)d_in[5];
    float* out = (float*)d_out;

    // 32768 (b,s) pairs, one wave32 each, 8 waves per 256-thread block.
    dim3 grid(BB * SS / 8);
    dim3 block(256);
    tokenizer_kernel<<<grid, block, 0, stream>>>(
        x_seq, cls_token, W, Bias, cat_emb, pos, out);
}